// ShortMemorySoftBinaryTreeNode_31834297598165
// MI455X (gfx1250) — compile-verified
//
#include <hip/hip_runtime.h>

// Soft binary tree, fully fused on gfx1250 (MI455X), fp32 WMMA path.
//
// Per-depth batched GEMM (16x128 @ 128x16 per node per wave) on
// v_wmma_f32_16x16x4_f32; parent/grandparent rank-1 terms folded in as one
// extra WMMA K-step (features 128/129 of EFF_IN=130). Recursion state (scores,
// in-place path-prob expansion) lives in LDS. Weights are software-pipelined:
// register-prefetch node n+1 while computing node n, double-buffered LDS
// staging, global_prefetch for node n+2. Epilogue pp @ leaf_logits is also WMMA.
//
// WMMA f32 16x16x4 layouts (CDNA5 ISA 7.12.2, wave32):
//  A 16x4 : lanes 0-15 -> M=lane, {VGPR0,VGPR1}={K0,K1}; lanes 16-31 -> {K2,K3}
//  B 4x16 : VGPR0: lanes0-15 = row K0 (N=lane), lanes16-31 = row K2; VGPR1: K1/K3
//  C/D    : lanes0-15 N=lane, M=vgpr(0..7); lanes16-31 N=lane-16, M=vgpr+8

#define INPUT_DIM   128
#define HID         16
#define MAX_DEPTH   8
#define N_INTERNAL  255
#define N_LEAVES    256
#define NUM_CLASSES 10
#define EFF_IN      130
#define WFLOATS     (EFF_IN * HID)      // 2080 floats per node
#define ROWS_PER_WG 64
#define THREADS     128                 // 4 waves of 32

typedef __attribute__((ext_vector_type(2))) float v2f;
typedef __attribute__((ext_vector_type(8))) float v8f;

struct SMem {
  float W[2][WFLOATS];                  // double-buffered W1 node block (130x16)
  float B1[2][HID];
  float W2[2][HID];
  float B2[2];
  float leafP[N_LEAVES * 16];           // leaf_logits padded 256x16 (cols 10..15 = 0)
  float PP[ROWS_PER_WG][N_LEAVES + 1];  // slot(node i, depth d) = i << (8-d); +1 pad
  float Cur[ROWS_PER_WG][129];          // scores at depth d    (padded rows)
  float Prev[ROWS_PER_WG][65];          // scores at depth d-1
  float Prev2[ROWS_PER_WG][33];         // scores at depth d-2
};

__global__ __launch_bounds__(THREADS) void tree_wmma_kernel(
    const float* __restrict__ x, const float* __restrict__ path_prob,
    const float* __restrict__ W1, const float* __restrict__ b1,
    const float* __restrict__ w2, const float* __restrict__ b2,
    const float* __restrict__ leaf_logits, float* __restrict__ out)
{
  __shared__ SMem sm;
  const int tid  = threadIdx.x;
  const int wave = tid >> 5;
  const int lane = tid & 31;
  const int half = lane >> 4;   // 0: lanes 0-15, 1: lanes 16-31
  const int col  = lane & 15;   // B/C column; also A-row index within wave tile
  const int kb   = half * 2;    // K sub-offset for this half
  const int wgRow0   = blockIdx.x * ROWS_PER_WG;
  const int waveRow0 = wgRow0 + wave * 16;
  const int rl       = wave * 16 + col;      // local row this lane owns for A

  // ---- cache this wave's 16x128 A-tile of x in registers (WMMA A layout) ----
  v2f afrag[32];
  {
    const v2f* x2 = (const v2f*)(x + (size_t)(waveRow0 + col) * INPUT_DIM);
    #pragma unroll
    for (int k = 0; k < 32; ++k) afrag[k] = x2[2 * k + half];  // K = 4k + 2*half, +1
  }

  // ---- init LDS state ----
  for (int j = tid; j < N_LEAVES * 16; j += THREADS) {
    const int lr = j >> 4, c = j & 15;
    sm.leafP[j] = (c < NUM_CLASSES) ? leaf_logits[lr * NUM_CLASSES + c] : 0.0f;
  }
  if (tid < ROWS_PER_WG) {
    sm.PP[tid][0]    = path_prob[wgRow0 + tid];
    sm.Prev[tid][0]  = 0.0f;
    sm.Prev2[tid][0] = 0.0f;
  }

  // ---- weight pipeline registers ----
  float4 wreg[5];
  float  b1v = 0.0f, w2v = 0.0f, b2v = 0.0f;

  // prologue: load node 0 and commit to buffer 0
  {
    const float4* src = (const float4*)(W1);
    #pragma unroll
    for (int t = 0; t < 4; ++t) wreg[t] = src[tid + THREADS * t];
    if (tid < 8)   wreg[4] = src[tid + 512];          // 520 float4 total
    if (tid < HID) { b1v = b1[tid]; w2v = w2[tid]; }
    if (tid == 0)  b2v = b2[0];

    float4* dst = (float4*)sm.W[0];
    #pragma unroll
    for (int t = 0; t < 4; ++t) dst[tid + THREADS * t] = wreg[t];
    if (tid < 8)   dst[tid + 512] = wreg[4];
    if (tid < HID) { sm.B1[0][tid] = b1v; sm.W2[0][tid] = w2v; }
    if (tid == 0)  sm.B2[0] = b2v;
  }
  __syncthreads();

  int curbuf = 0;
  for (int d = 0; d < MAX_DEPTH; ++d) {
    const int nNodes   = 1 << d;
    const int nodeBase = nNodes - 1;

    for (int i = 0; i < nNodes; ++i) {
      const int n = nodeBase + i;

      // -- issue register prefetch of node n+1 (latency hidden by WMMAs below) --
      const bool havenext = (n + 1 < N_INTERNAL);
      if (havenext) {
        const float4* src = (const float4*)(W1 + (size_t)(n + 1) * WFLOATS);
        #pragma unroll
        for (int t = 0; t < 4; ++t) wreg[t] = src[tid + THREADS * t];
        if (tid < 8)   wreg[4] = src[tid + 512];
        if (tid < HID) { b1v = b1[(n + 1) * HID + tid]; w2v = w2[(n + 1) * HID + tid]; }
        if (tid == 0)  b2v = b2[n + 1];
      }
      if (n + 2 < N_INTERNAL)   // warm L2/L0 for the node after that
        __builtin_prefetch(W1 + (size_t)(n + 2) * WFLOATS + tid * 16, 0, 1);

      // ---- 16x16 hidden block: Xtile(16x128) @ W1[n](128x16), 32 WMMA steps ----
      const float* Wc = sm.W[curbuf];
      v8f acc0 = {0.f,0.f,0.f,0.f,0.f,0.f,0.f,0.f};
      v8f acc1 = {0.f,0.f,0.f,0.f,0.f,0.f,0.f,0.f};
      #pragma unroll
      for (int k = 0; k < 32; k += 2) {
        v2f bf0, bf1;
        bf0.x = Wc[(4 * k + kb) * HID + col];
        bf0.y = Wc[(4 * k + kb + 1) * HID + col];
        bf1.x = Wc[(4 * (k + 1) + kb) * HID + col];
        bf1.y = Wc[(4 * (k + 1) + kb + 1) * HID + col];
        acc0 = __builtin_amdgcn_wmma_f32_16x16x4_f32(false, afrag[k],     false, bf0,
                                                     (short)0, acc0, false, false);
        acc1 = __builtin_amdgcn_wmma_f32_16x16x4_f32(false, afrag[k + 1], false, bf1,
                                                     (short)0, acc1, false, false);
      }
      // extra K-step: parent (feature 128) + grandparent (feature 129); K=130,131 -> 0
      {
        v2f ae = {0.f, 0.f}, be = {0.f, 0.f};
        if (half == 0) {
          ae.x = sm.Prev[rl][i >> 1];
          ae.y = sm.Prev2[rl][i >> 2];
          be.x = Wc[128 * HID + col];
          be.y = Wc[129 * HID + col];
        }
        acc0 = __builtin_amdgcn_wmma_f32_16x16x4_f32(false, ae, false, be,
                                                     (short)0, acc0, false, false);
      }

      // ---- t = relu(h + b1); score = sum_hid t*w2 + b2 (16-lane reduce) ----
      const float b1c = sm.B1[curbuf][col];
      const float w2c = sm.W2[curbuf][col];
      const float b2c = sm.B2[curbuf];
      #pragma unroll
      for (int r = 0; r < 8; ++r) {
        float h = acc0[r] + acc1[r];
        float t = fmaxf(h + b1c, 0.0f);
        float v = t * w2c;
        v += __shfl_xor(v, 1);
        v += __shfl_xor(v, 2);
        v += __shfl_xor(v, 4);
        v += __shfl_xor(v, 8);          // stays within each 16-lane half
        if (col == 0) sm.Cur[wave * 16 + half * 8 + r][i] = v + b2c;
      }

      // -- commit prefetched node n+1 into the other buffer, flip --
      if (havenext) {
        float4* dst = (float4*)sm.W[curbuf ^ 1];
        #pragma unroll
        for (int t = 0; t < 4; ++t) dst[tid + THREADS * t] = wreg[t];
        if (tid < 8)   dst[tid + 512] = wreg[4];
        if (tid < HID) { sm.B1[curbuf ^ 1][tid] = b1v; sm.W2[curbuf ^ 1][tid] = w2v; }
        if (tid == 0)  sm.B2[curbuf ^ 1] = b2v;
      }
      __syncthreads();
      curbuf ^= 1;
    } // node loop

    // ---- pp expansion: pp[2i]=pp*p, pp[2i+1]=pp*(1-p); in-place slot trick ----
    {
      const int shift = MAX_DEPTH - d;
      const int gap   = 1 << (shift - 1);
      for (int tix = tid; tix < (ROWS_PER_WG << d); tix += THREADS) {
        const int r = tix >> d;
        const int i = tix & (nNodes - 1);
        const float s = sm.Cur[r][i];
        const float p = 1.0f / (1.0f + expf(-s));   // TEMP = 1.0
        const int slot = i << shift;
        const float v = sm.PP[r][slot];
        sm.PP[r][slot]       = v * p;
        sm.PP[r][slot + gap] = v * (1.0f - p);
      }
    }
    __syncthreads();
    // ---- rotate score history: Prev2 <- Prev, Prev <- Cur ----
    if (d < MAX_DEPTH - 1) {
      if (d >= 1) {
        for (int tix = tid; tix < (ROWS_PER_WG << (d - 1)); tix += THREADS) {
          const int r = tix >> (d - 1);
          const int j = tix & ((1 << (d - 1)) - 1);
          sm.Prev2[r][j] = sm.Prev[r][j];
        }
      }
      __syncthreads();
      for (int tix = tid; tix < (ROWS_PER_WG << d); tix += THREADS) {
        const int r = tix >> d;
        const int j = tix & (nNodes - 1);
        sm.Prev[r][j] = sm.Cur[r][j];
      }
      __syncthreads();
    }
  } // depth loop

  __syncthreads();
  // ---- epilogue as WMMA: out(16x10) = PP(16x256) @ leafP(256x16), per wave ----
  {
    v8f e0 = {0.f,0.f,0.f,0.f,0.f,0.f,0.f,0.f};
    v8f e1 = {0.f,0.f,0.f,0.f,0.f,0.f,0.f,0.f};
    #pragma unroll
    for (int k = 0; k < 64; k += 2) {
      v2f a0, bq0, a1, bq1;
      a0.x  = sm.PP[rl][4 * k + kb];
      a0.y  = sm.PP[rl][4 * k + kb + 1];
      bq0.x = sm.leafP[(4 * k + kb) * 16 + col];
      bq0.y = sm.leafP[(4 * k + kb + 1) * 16 + col];
      a1.x  = sm.PP[rl][4 * (k + 1) + kb];
      a1.y  = sm.PP[rl][4 * (k + 1) + kb + 1];
      bq1.x = sm.leafP[(4 * (k + 1) + kb) * 16 + col];
      bq1.y = sm.leafP[(4 * (k + 1) + kb + 1) * 16 + col];
      e0 = __builtin_amdgcn_wmma_f32_16x16x4_f32(false, a0, false, bq0,
                                                 (short)0, e0, false, false);
      e1 = __builtin_amdgcn_wmma_f32_16x16x4_f32(false, a1, false, bq1,
                                                 (short)0, e1, false, false);
    }
    if (col < NUM_CLASSES) {
      #pragma unroll
      for (int r = 0; r < 8; ++r)
        out[(size_t)(waveRow0 + half * 8 + r) * NUM_CLASSES + col] = e0[r] + e1[r];
    }
  }
}

extern "C" void kernel_launch(void* const* d_in, const int* in_sizes, int n_in,
                              void* d_out, int out_size, void* d_ws, size_t ws_size,
                              hipStream_t stream) {
  const float* x    = (const float*)d_in[0];
  const float* pp   = (const float*)d_in[1];
  const float* W1   = (const float*)d_in[2];
  const float* b1   = (const float*)d_in[3];
  const float* w2   = (const float*)d_in[4];
  const float* b2   = (const float*)d_in[5];
  const float* leaf = (const float*)d_in[6];
  float* out = (float*)d_out;

  const int B = in_sizes[0] / INPUT_DIM;      // 32768
  dim3 grid(B / ROWS_PER_WG);                 // 512 workgroups
  dim3 block(THREADS);                        // 128 threads = 4 waves
  tree_wmma_kernel<<<grid, block, 0, stream>>>(x, pp, W1, b1, w2, b2, leaf, out);
}